// EdgeConvBlockV2_20770461843674
// MI455X (gfx1250) — compile-verified
//
#include <hip/hip_runtime.h>
#include <hip/hip_bf16.h>

typedef __attribute__((ext_vector_type(2))) float v2f;
typedef __attribute__((ext_vector_type(8))) float v8f;

#define BATCH 8
#define CH    64
#define NPTS  4096
#define KNN   20
#define BN_EPS 1e-5f

// ---------------------------------------------------------------------------
// Kernel 1: transpose features to (B,N,C) rows, compute ||x||^2, and the two
// 1x1 convs  s = (w1+w2) x  and  e = w2 x  (weights staged in LDS).
// ---------------------------------------------------------------------------
__global__ __launch_bounds__(256) void ec_prep_kernel(
    const float* __restrict__ f,   // (B,C,N)
    const float* __restrict__ w1,  // (O,C)
    const float* __restrict__ w2,  // (O,C)
    float* __restrict__ xT,        // (B,N,C)
    float* __restrict__ sT,        // (B,N,C)
    float* __restrict__ eT,        // (B,N,C)
    float* __restrict__ sq)        // (B,N)
{
    __shared__ float w12s[CH * CH];
    __shared__ float w2s[CH * CH];
    int tid = threadIdx.x;
    for (int i = tid; i < CH * CH; i += 256) {
        float a = w1[i], b = w2[i];
        w12s[i] = a + b;
        w2s[i] = b;
    }
    __syncthreads();

    int gid = blockIdx.x * 256 + tid;       // 0 .. B*N-1
    int b = gid >> 12;
    int n = gid & (NPTS - 1);
    const float* fb = f + (size_t)b * CH * NPTS;

    float x[CH];
    float s2 = 0.0f;
#pragma unroll
    for (int c = 0; c < CH; ++c) {
        float v = fb[(size_t)c * NPTS + n];
        x[c] = v;
        s2 = fmaf(v, v, s2);
    }
    sq[gid] = s2;

    float* xr = xT + (size_t)gid * CH;
#pragma unroll
    for (int c = 0; c < CH; ++c) xr[c] = x[c];

    float* sr = sT + (size_t)gid * CH;
    float* er = eT + (size_t)gid * CH;
    for (int o = 0; o < CH; ++o) {
        const float* wr12 = &w12s[o * CH];
        const float* wr2  = &w2s[o * CH];
        float ss = 0.0f, ee = 0.0f;
#pragma unroll
        for (int c = 0; c < CH; ++c) {
            ss = fmaf(wr12[c], x[c], ss);
            ee = fmaf(wr2[c], x[c], ee);
        }
        sr[o] = ss;
        er[o] = ee;
    }
}

// ---------------------------------------------------------------------------
// Kernel 2: fused pairwise-distance GEMM (WMMA f32 16x16x4, K=64) + top-20.
// One wave32 per 16 query rows; 16x16 distance tiles never leave LDS.
// ---------------------------------------------------------------------------
__global__ __launch_bounds__(32) void ec_knn_kernel(
    const float* __restrict__ xT,  // (B,N,C)
    const float* __restrict__ sq,  // (B,N)
    int* __restrict__ knn)         // (B,N,KNN)
{
    const int b    = blockIdx.y;
    const int m0   = blockIdx.x * 16;
    const int lane = threadIdx.x;
    const int ml   = lane & 15;       // row (A) / col (B) within tile
    const int kb2  = lane >> 4;       // 0 -> K pair {0,1}, 1 -> K pair {2,3}

    const float* xb  = xT + (size_t)b * NPTS * CH;
    const float* sqb = sq + (size_t)b * NPTS;

    // A fragments: query rows, kept in registers for the whole sweep.
    v2f afrag[16];
    {
        const v2f* qrow = (const v2f*)(xb + (size_t)(m0 + ml) * CH);
#pragma unroll
        for (int s = 0; s < 16; ++s) afrag[s] = qrow[2 * s + kb2];
    }

    float best_d[KNN];
    int   best_i[KNN];
#pragma unroll
    for (int t = 0; t < KNN; ++t) { best_d[t] = __builtin_inff(); best_i[t] = 0; }
    float cur_worst = __builtin_inff();

    __shared__ float dist[16 * 17];   // +1 pad -> conflict-free column scan

    for (int n0 = 0; n0 < NPTS; n0 += 16) {
        if (n0 + 16 < NPTS)
            __builtin_prefetch(xb + (size_t)(n0 + 16 + ml) * CH, 0, 0);

        // B fragments: column-point rows (B[k][n] = x[n][c+k]).
        const v2f* crow = (const v2f*)(xb + (size_t)(n0 + ml) * CH);
        v2f bfrag[16];
#pragma unroll
        for (int s = 0; s < 16; ++s) bfrag[s] = crow[2 * s + kb2];

        v8f acc = {0.f, 0.f, 0.f, 0.f, 0.f, 0.f, 0.f, 0.f};
#pragma unroll
        for (int s = 0; s < 16; ++s) {
            acc = __builtin_amdgcn_wmma_f32_16x16x4_f32(
                false, afrag[s], false, bfrag[s], (short)0, acc, false, false);
        }

        // distance (row-constant ||x_m||^2 dropped; ordering unaffected)
        const float sqc   = sqb[n0 + ml];
        const int   rbase = kb2 * 8;
#pragma unroll
        for (int v = 0; v < 8; ++v)
            dist[(rbase + v) * 17 + ml] = fmaf(-2.0f, acc[v], sqc);
        __syncthreads();

        if (lane < 16) {
            for (int j = 0; j < 16; ++j) {
                float d = dist[lane * 17 + j];
                if (d < cur_worst) {
                    int id = n0 + j;
                    // replace current worst entry (fully unrolled -> registers)
                    int wpos = 0; float w = best_d[0];
#pragma unroll
                    for (int t = 1; t < KNN; ++t)
                        if (best_d[t] > w) { w = best_d[t]; wpos = t; }
#pragma unroll
                    for (int t = 0; t < KNN; ++t)
                        if (t == wpos) { best_d[t] = d; best_i[t] = id; }
                    w = best_d[0];
#pragma unroll
                    for (int t = 1; t < KNN; ++t) w = fmaxf(w, best_d[t]);
                    cur_worst = w;
                }
            }
        }
        __syncthreads();
    }

    if (lane < 16) {
        int* kr = knn + ((size_t)b * NPTS + (m0 + lane)) * KNN;
#pragma unroll
        for (int t = 0; t < KNN; ++t) kr[t] = best_i[t];
    }
}

// ---------------------------------------------------------------------------
// Kernel 3: per-block partial BN statistics of ef = s[pt] - e[nbr] per channel.
// Deterministic (no atomics): 512 blocks x 64 channels of partials.
// ---------------------------------------------------------------------------
__global__ __launch_bounds__(256) void ec_stats_kernel(
    const float* __restrict__ sT, const float* __restrict__ eT,
    const int* __restrict__ knn,
    float* __restrict__ psum, float* __restrict__ psumsq)
{
    const int c  = threadIdx.x & 63;
    const int p4 = threadIdx.x >> 6;           // 0..3
    const int base_pt = blockIdx.x * 64;       // 64 points per block
    const int b = base_pt >> 12;

    float sum = 0.0f, ssq = 0.0f;
    for (int pp = p4; pp < 64; pp += 4) {
        const int pt = base_pt + pp;
        const float sv = sT[(size_t)pt * CH + c];
        const int* id = knn + (size_t)pt * KNN;
#pragma unroll
        for (int j = 0; j < KNN; ++j) {
            int nb = id[j];
            float ev = eT[((size_t)(b << 12) + nb) * CH + c];
            float d = sv - ev;
            sum += d;
            ssq = fmaf(d, d, ssq);
        }
    }

    __shared__ float lsum[256];
    __shared__ float lssq[256];
    lsum[threadIdx.x] = sum;
    lssq[threadIdx.x] = ssq;
    __syncthreads();
    if (threadIdx.x < 64) {
        // fixed summation order -> deterministic
        float a = lsum[c] + lsum[c + 64] + lsum[c + 128] + lsum[c + 192];
        float q = lssq[c] + lssq[c + 64] + lssq[c + 128] + lssq[c + 192];
        psum[blockIdx.x * 64 + c]   = a;
        psumsq[blockIdx.x * 64 + c] = q;
    }
}

// ---------------------------------------------------------------------------
// Kernel 4: reduce partials -> scale/shift per channel.
// ---------------------------------------------------------------------------
__global__ __launch_bounds__(64) void ec_bn_kernel(
    const float* __restrict__ psum, const float* __restrict__ psumsq,
    const float* __restrict__ gamma, const float* __restrict__ beta,
    float* __restrict__ scaleshift)
{
    const int c = threadIdx.x;
    float s = 0.0f, q = 0.0f;
    for (int blk = 0; blk < 512; ++blk) {
        s += psum[blk * 64 + c];
        q += psumsq[blk * 64 + c];
    }
    const float inv = 1.0f / ((float)BATCH * (float)NPTS * (float)KNN);
    float mean = s * inv;
    float var = fmaf(-mean, mean, q * inv);
    float sc = gamma[c] * rsqrtf(var + BN_EPS);
    scaleshift[c]      = sc;
    scaleshift[64 + c] = fmaf(-mean, sc, beta[c]);
}

// ---------------------------------------------------------------------------
// Kernel 5: fused gather + BN + ReLU + max over k; LDS transpose for
// coalesced (B,O,N) output writes.
// ---------------------------------------------------------------------------
__global__ __launch_bounds__(256) void ec_final_kernel(
    const float* __restrict__ sT, const float* __restrict__ eT,
    const int* __restrict__ knn, const float* __restrict__ scaleshift,
    float* __restrict__ out)   // (B,O,N)
{
    __shared__ float tile[64 * 65];
    const int c  = threadIdx.x & 63;
    const int p4 = threadIdx.x >> 6;
    const int base_pt = blockIdx.x * 64;
    const int b = base_pt >> 12;
    const float sc = scaleshift[c];
    const float sh = scaleshift[64 + c];

    for (int pp = p4; pp < 64; pp += 4) {
        const int pt = base_pt + pp;
        const float sv = sT[(size_t)pt * CH + c];
        const int* id = knn + (size_t)pt * KNN;
        float m = 0.0f;   // max_j relu(v_j) == max(0, max_j v_j)
#pragma unroll
        for (int j = 0; j < KNN; ++j) {
            int nb = id[j];
            float ev = eT[((size_t)(b << 12) + nb) * CH + c];
            float v = fmaf(sc, sv - ev, sh);
            m = fmaxf(m, v);
        }
        tile[pp * 65 + c] = m;
    }
    __syncthreads();

    const int n0 = base_pt & (NPTS - 1);
    for (int o4 = 0; o4 < 64; o4 += 4) {
        int o = o4 + p4;
        out[(((size_t)b * CH + o) << 12) + n0 + c] = tile[c * 65 + o];
    }
}

// ---------------------------------------------------------------------------
extern "C" void kernel_launch(void* const* d_in, const int* in_sizes, int n_in,
                              void* d_out, int out_size, void* d_ws, size_t ws_size,
                              hipStream_t stream) {
    const float* features = (const float*)d_in[0];
    const float* w1       = (const float*)d_in[1];
    const float* w2       = (const float*)d_in[2];
    const float* gamma    = (const float*)d_in[3];
    const float* beta     = (const float*)d_in[4];
    // d_in[5] = k (always 20; hardcoded)

    char* ws = (char*)d_ws;
    const size_t BN  = (size_t)BATCH * NPTS;        // 32768
    float* xT  = (float*)(ws);                       //  8 MB
    float* sT  = (float*)(ws + BN * CH * 4);         //  8 MB
    float* eT  = (float*)(ws + 2 * BN * CH * 4);     //  8 MB
    float* sq  = (float*)(ws + 3 * BN * CH * 4);     // 128 KB
    int*   knn = (int*)  (ws + 3 * BN * CH * 4 + BN * 4);            // 2.5 MB
    float* psum   = (float*)(ws + 3 * BN * CH * 4 + BN * 4 + BN * KNN * 4);
    float* psumsq = psum + 512 * 64;
    float* scsh   = psumsq + 512 * 64;

    ec_prep_kernel<<<dim3((int)(BN / 256)), 256, 0, stream>>>(
        features, w1, w2, xT, sT, eT, sq);

    ec_knn_kernel<<<dim3(NPTS / 16, BATCH), 32, 0, stream>>>(xT, sq, knn);

    ec_stats_kernel<<<dim3((int)(BN / 64)), 256, 0, stream>>>(
        sT, eT, knn, psum, psumsq);

    ec_bn_kernel<<<1, 64, 0, stream>>>(psum, psumsq, gamma, beta, scsh);

    ec_final_kernel<<<dim3((int)(BN / 64)), 256, 0, stream>>>(
        sT, eT, knn, scsh, (float*)d_out);
}